// GNNLayer_19739669692578
// MI455X (gfx1250) — compile-verified
//
#include <hip/hip_runtime.h>

typedef __attribute__((ext_vector_type(2))) float v2f;
typedef __attribute__((ext_vector_type(8))) float v8f;

#define N_NODES 10000
#define N_EDGES 160000
#define F       128
#define BATCH   8

// ---------------------------------------------------------------------------
// 1) degree count (float atomics; degree shared across batch)
// ---------------------------------------------------------------------------
__global__ void deg_kernel(const long long* __restrict__ edge_index,
                           float* __restrict__ deg) {
  int e = blockIdx.x * blockDim.x + threadIdx.x;
  if (e < N_EDGES) {
    int dst = (int)edge_index[N_EDGES + e];
    atomicAdd(&deg[dst], 1.0f);
  }
}

// 2) deg <- 1 / max(deg, 1) in place
__global__ void invdeg_kernel(float* __restrict__ deg) {
  int n = blockIdx.x * blockDim.x + threadIdx.x;
  if (n < N_NODES) {
    deg[n] = 1.0f / fmaxf(deg[n], 1.0f);
  }
}

// ---------------------------------------------------------------------------
// 3) neighbor-sum scatter: one block per edge, 256 thr = 8 batches x 32 f4
//    coalesced float4 gather of x[b,src,:], atomic adds into out[b,dst,:]
// ---------------------------------------------------------------------------
__global__ void __launch_bounds__(256) scatter_kernel(
    const float* __restrict__ x, const long long* __restrict__ edge_index,
    float* __restrict__ out) {
  unsigned tid = blockIdx.x * 256u + threadIdx.x;
  unsigned f4 = tid & 31u;        // feature chunk of 4 floats
  unsigned b  = (tid >> 5) & 7u;  // batch
  unsigned e  = tid >> 8;         // edge
  int src = (int)edge_index[e];
  int dst = (int)edge_index[N_EDGES + e];
  const float4 v =
      *(const float4*)(x + ((size_t)b * N_NODES + src) * F + f4 * 4u);
  float* o = out + ((size_t)b * N_NODES + dst) * F + f4 * 4u;
  atomicAdd(o + 0, v.x);
  atomicAdd(o + 1, v.y);
  atomicAdd(o + 2, v.z);
  atomicAdd(o + 3, v.w);
}

// ---------------------------------------------------------------------------
// 4) fused dual GEMM with V_WMMA_F32_16X16X4_F32 (exact fp32):
//    out[r, o] = sum_f (sum_nbr[r,f]*invdeg) * Wl[o,f]
//              + sum_f x[r,f]               * Wr[o,f]  + bias[o]
//    Wave owns one 16-row tile (10000 % 16 == 0 per batch -> no straddle),
//    8 column tiles (full 128 outputs) accumulated in registers.
//    Neighbor sums are read from `out` and only overwritten after all of
//    this wave's A reads complete (rows are wave-private).
// ---------------------------------------------------------------------------
__global__ void __launch_bounds__(256) sage_wmma_kernel(
    const float* __restrict__ x, const float* __restrict__ Wl,
    const float* __restrict__ bias, const float* __restrict__ Wr,
    const float* __restrict__ invdeg, float* __restrict__ out) {
  const int lane = threadIdx.x & 31;
  const int wave = threadIdx.x >> 5;
  const int rowTile = blockIdx.x * 8 + wave;
  const int row0 = rowTile * 16;
  if (row0 >= BATCH * N_NODES) return;

  const int m    = lane & 15;   // A-row / B-col within tile (ISA layout)
  const int half = lane >> 4;   // K phase: VGPR v holds K = 2*half + v
  const int r    = row0 + m;    // this lane's A row
  const int node = r % N_NODES; // tiles never straddle batches
  const float inv_d = invdeg[node];

  const float* __restrict__ sumRow = out + (size_t)r * F;
  const float* __restrict__ xRow   = x   + (size_t)r * F;

  v8f acc[8] = {};  // 8 N-tiles x (16x16 f32 / 32 lanes = 8 VGPRs)

  for (int k = 0; k < F; k += 4) {
    const int kk = k + half * 2;
    v2f a_m, a_x;
    {
      const float2 t = *(const float2*)(sumRow + kk);
      a_m.x = t.x * inv_d;  // fuse mean = sum / deg into A fragment
      a_m.y = t.y * inv_d;
    }
    {
      const float2 t = *(const float2*)(xRow + kk);
      a_x.x = t.x;
      a_x.y = t.y;
    }
#pragma unroll
    for (int ot = 0; ot < 8; ++ot) {
      const int n = ot * 16 + m;  // B[k][n] = W[n][k] -> contiguous float2
      const float2 tl = *(const float2*)(Wl + (size_t)n * F + kk);
      const float2 tr = *(const float2*)(Wr + (size_t)n * F + kk);
      v2f b_l, b_r;
      b_l.x = tl.x; b_l.y = tl.y;
      b_r.x = tr.x; b_r.y = tr.y;
      acc[ot] = __builtin_amdgcn_wmma_f32_16x16x4_f32(
          false, a_m, false, b_l, (short)0, acc[ot], false, false);
      acc[ot] = __builtin_amdgcn_wmma_f32_16x16x4_f32(
          false, a_x, false, b_r, (short)0, acc[ot], false, false);
    }
  }

  // C/D layout: VGPR v -> M = v + 8*half, N = m (per lane); fuse bias add.
#pragma unroll
  for (int ot = 0; ot < 8; ++ot) {
    const int col = ot * 16 + m;
    const float bv = bias[col];
#pragma unroll
    for (int v = 0; v < 8; ++v) {
      const int rr = row0 + v + half * 8;
      out[(size_t)rr * F + col] = acc[ot][v] + bv;
    }
  }
}

// ---------------------------------------------------------------------------
extern "C" void kernel_launch(void* const* d_in, const int* in_sizes, int n_in,
                              void* d_out, int out_size, void* d_ws,
                              size_t ws_size, hipStream_t stream) {
  const float* x      = (const float*)d_in[0];
  const float* Wl     = (const float*)d_in[1];
  const float* bl     = (const float*)d_in[2];
  const float* Wr     = (const float*)d_in[3];
  const long long* ei = (const long long*)d_in[4];  // int64 edge_index [2,E]
  float* out = (float*)d_out;
  float* deg = (float*)d_ws;  // 10000 floats of scratch

  hipMemsetAsync(deg, 0, N_NODES * sizeof(float), stream);
  hipMemsetAsync(out, 0, (size_t)out_size * sizeof(float), stream);

  deg_kernel<<<(N_EDGES + 255) / 256, 256, 0, stream>>>(ei, deg);
  invdeg_kernel<<<(N_NODES + 255) / 256, 256, 0, stream>>>(deg);
  // E blocks: each handles one edge across 8 batches x 32 float4 chunks
  scatter_kernel<<<N_EDGES, 256, 0, stream>>>(x, ei, out);
  // 80000 rows / 16 per wave / 8 waves per block = 625 blocks
  sage_wmma_kernel<<<625, 256, 0, stream>>>(x, Wl, bl, Wr, deg, out);
}